// GNNRegressor_50079318671678
// MI455X (gfx1250) — compile-verified
//
#include <hip/hip_runtime.h>
#include <stdint.h>

#define N_NODES  80000
#define N_EDGES  1280000
#define IN_CH    128
#define HID      64
#define N_GRAPHS 64
#define KEEP_P   0.7f
#define INV_KEEP (1.0f/0.7f)

typedef __attribute__((ext_vector_type(2))) float v2f;
typedef __attribute__((ext_vector_type(8))) float v8f;

// ---------------------------------------------------------------- RNG (stateless)
__device__ __forceinline__ float urand01(uint32_t a, uint32_t b) {
  uint32_t x = a * 0x9E3779B1u + b * 0x85EBCA77u;
  x ^= x >> 16; x *= 0x7FEB352Du;
  x ^= x >> 15; x *= 0x846CA68Bu;
  x ^= x >> 16;
  return (float)(x >> 8) * (1.0f/16777216.0f);
}

// ---------------------------------------------------------------- utility
__global__ void zero_f32(float* __restrict__ p, int n) {
  int i = blockIdx.x * blockDim.x + threadIdx.x;
  if (i < n) p[i] = 0.0f;
}

// deg[d] += 1 per edge
__global__ void deg_kernel(const int* __restrict__ dst, float* __restrict__ deg) {
  int e = blockIdx.x * blockDim.x + threadIdx.x;
  if (e < N_EDGES) atomicAdd(&deg[dst[e]], 1.0f);
}

// in-place: deg -> rsqrt(deg + 1)   (self-loop +1, PyG norm)
__global__ void dinv_kernel(float* __restrict__ deg) {
  int i = blockIdx.x * blockDim.x + threadIdx.x;
  if (i < N_NODES) deg[i] = rsqrtf(deg[i] + 1.0f);
}

// ---------------------------------------------------------------- fp32 WMMA GEMM
// out[M=80000][64] = A[M][K] * W[K][64]   (no bias: GCN adds bias post-aggregation)
// block = 256 thr = 8 waves; block tile 32x64; wave tile 16x16; K-step 4 (f32 WMMA)
template<int K>
__global__ __launch_bounds__(256) void gemm_wmma(const float* __restrict__ A,
                                                 const float* __restrict__ W,
                                                 float* __restrict__ out) {
  __shared__ float sx [32 * (K + 2)];   // A tile, row-major, padded stride
  __shared__ float swt[64 * (K + 2)];   // W transposed: swt[n][k]

  const int t  = threadIdx.x;
  const int m0 = blockIdx.x * 32;

  // stage A tile: global is contiguous (rows m0..m0+31), fully coalesced
  for (int i = t; i < 32 * K; i += 256)
    sx[(i / K) * (K + 2) + (i % K)] = A[(size_t)m0 * K + i];
  // stage W transposed (coalesced reads)
  for (int i = t; i < K * 64; i += 256) {
    int k = i >> 6, n = i & 63;
    swt[n * (K + 2) + k] = W[i];
  }
  __syncthreads();

  const int lane = t & 31;
  const int wv   = t >> 5;          // wave 0..7
  const int mt   = wv >> 2;         // M tile 0..1
  const int nt   = wv & 3;          // N tile 0..3
  const int half = lane >> 4;       // 0: lanes 0-15, 1: lanes 16-31
  const int r16  = lane & 15;

  const float* ax = &sx [(mt * 16 + r16) * (K + 2)];  // A row M = r16
  const float* bx = &swt[(nt * 16 + r16) * (K + 2)];  // B col N = r16 (from W^T)

  v8f c = {0.f, 0.f, 0.f, 0.f, 0.f, 0.f, 0.f, 0.f};
  #pragma unroll
  for (int k0 = 0; k0 < K; k0 += 4) {
    // f32 A 16x4 layout: VGPR0 = {K0 | K2}, VGPR1 = {K1 | K3} across lane halves
    const float2 af = *(const float2*)(ax + k0 + 2 * half);
    const float2 bf = *(const float2*)(bx + k0 + 2 * half);
    v2f a; a.x = af.x; a.y = af.y;
    v2f b; b.x = bf.x; b.y = bf.y;
    c = __builtin_amdgcn_wmma_f32_16x16x4_f32(false, a, false, b,
                                              (short)0, c, false, false);
  }

  // C/D 16x16 f32 layout: VGPR r -> M = r + 8*half, N = r16
  const int orow = m0 + mt * 16 + 8 * half;
  const int ocol = nt * 16 + r16;
  #pragma unroll
  for (int r = 0; r < 8; ++r)
    out[(size_t)(orow + r) * HID + ocol] = c[r];
}

// ---------------------------------------------------------------- edge scatter
// one wave per edge: src/dst/norm uniform across the wave, float2 per lane
__global__ __launch_bounds__(256) void scatter_kernel(const int* __restrict__ src,
                                                      const int* __restrict__ dst,
                                                      const float* __restrict__ dinv,
                                                      const float* __restrict__ h,
                                                      float* __restrict__ agg) {
  int gid  = blockIdx.x * blockDim.x + threadIdx.x;
  int e    = gid >> 5;
  int lane = gid & 31;
  if (e >= N_EDGES) return;
  int s = src[e], d = dst[e];
  float wgt = dinv[s] * dinv[d];
  const float2 hv = *(const float2*)&h[(size_t)s * HID + lane * 2];
  float* ap = &agg[(size_t)d * HID + lane * 2];
  atomicAdd(ap,     hv.x * wgt);
  atomicAdd(ap + 1, hv.y * wgt);
}

// out = dropout(relu(agg + h*dinv^2 + bias))
__global__ void post_kernel(const float* __restrict__ agg, const float* __restrict__ h,
                            const float* __restrict__ dinv, const float* __restrict__ bias,
                            float* __restrict__ out, uint32_t seed) {
  int idx = blockIdx.x * blockDim.x + threadIdx.x;
  if (idx >= N_NODES * HID) return;
  int i = idx >> 6, cch = idx & 63;
  float di = dinv[i];
  float v = agg[idx] + h[idx] * di * di + bias[cch];
  v = fmaxf(v, 0.0f);
  float u = urand01(seed, (uint32_t)idx);
  out[idx] = (u < KEEP_P) ? v * INV_KEEP : 0.0f;
}

// global mean pool accumulation
__global__ void pool_kernel(const float* __restrict__ h, const int* __restrict__ batch,
                            float* __restrict__ pooled, float* __restrict__ cnt) {
  int idx = blockIdx.x * blockDim.x + threadIdx.x;
  if (idx >= N_NODES * HID) return;
  int i = idx >> 6, cch = idx & 63;
  int g = batch[i];
  atomicAdd(&pooled[g * HID + cch], h[idx]);
  if (cch == 0) atomicAdd(&cnt[g], 1.0f);
}

// MLP head: mean -> Linear(64,64) -> ReLU -> Dropout -> Linear(64,1)
__global__ __launch_bounds__(256) void head_kernel(const float* __restrict__ pooled,
                                                   const float* __restrict__ cnt,
                                                   const float* __restrict__ Wm1,
                                                   const float* __restrict__ bm1,
                                                   const float* __restrict__ Wm2,
                                                   const float* __restrict__ bm2,
                                                   float* __restrict__ out,
                                                   uint32_t seed) {
  __shared__ float P [N_GRAPHS * HID];
  __shared__ float Mh[N_GRAPHS * HID];
  const int t = threadIdx.x;
  for (int i = t; i < N_GRAPHS * HID; i += 256) {
    int g = i >> 6;
    P[i] = pooled[i] / fmaxf(cnt[g], 1.0f);
  }
  __syncthreads();
  for (int i = t; i < N_GRAPHS * HID; i += 256) {
    int g = i >> 6, j = i & 63;
    float acc = bm1[j];
    #pragma unroll 8
    for (int k = 0; k < HID; ++k) acc += P[g * HID + k] * Wm1[k * HID + j];
    acc = fmaxf(acc, 0.0f);
    float u = urand01(seed, (uint32_t)i);
    Mh[i] = (u < KEEP_P) ? acc * INV_KEEP : 0.0f;
  }
  __syncthreads();
  if (t < N_GRAPHS) {
    float acc = bm2[0];
    #pragma unroll 8
    for (int j = 0; j < HID; ++j) acc += Mh[t * HID + j] * Wm2[j];
    out[t] = acc;
  }
}

// ---------------------------------------------------------------- launcher
static inline int cdiv(long a, long b) { return (int)((a + b - 1) / b); }

extern "C" void kernel_launch(void* const* d_in, const int* in_sizes, int n_in,
                              void* d_out, int out_size, void* d_ws, size_t ws_size,
                              hipStream_t stream) {
  (void)in_sizes; (void)n_in; (void)out_size; (void)ws_size;
  const float* x    = (const float*)d_in[0];
  const int*   ei   = (const int*)  d_in[1];
  const int*   bat  = (const int*)  d_in[2];
  const float* W1   = (const float*)d_in[3];
  const float* b1   = (const float*)d_in[4];
  const float* W2   = (const float*)d_in[5];
  const float* b2   = (const float*)d_in[6];
  const float* W3   = (const float*)d_in[7];
  const float* b3   = (const float*)d_in[8];
  const float* Wm1  = (const float*)d_in[9];
  const float* bm1  = (const float*)d_in[10];
  const float* Wm2  = (const float*)d_in[11];
  const float* bm2  = (const float*)d_in[12];
  const int* src = ei;
  const int* dst = ei + N_EDGES;
  float* out = (float*)d_out;

  // workspace carve (floats)
  float* ws   = (float*)d_ws;
  size_t o    = 0;
  float* dinv = ws + o; o += 80128;                    // deg -> dinv (in place)
  float* hA   = ws + o; o += (size_t)N_NODES * HID;    // GEMM output h = X W
  float* hB   = ws + o; o += (size_t)N_NODES * HID;    // post-activation output
  float* agg  = ws + o; o += (size_t)N_NODES * HID;    // neighbor aggregation
  float* pooled = ws + o; o += N_GRAPHS * HID;
  float* cnt    = ws + o; o += N_GRAPHS;

  const dim3 B(256);
  const int NH = N_NODES * HID;

  // degree / normalization
  zero_f32  <<<cdiv(N_NODES, 256), B, 0, stream>>>(dinv, N_NODES);
  deg_kernel<<<cdiv(N_EDGES, 256), B, 0, stream>>>(dst, dinv);
  dinv_kernel<<<cdiv(N_NODES, 256), B, 0, stream>>>(dinv);

  // ---- layer 1 (K = 128)
  gemm_wmma<IN_CH><<<N_NODES / 32, B, 0, stream>>>(x, W1, hA);
  zero_f32<<<cdiv(NH, 256), B, 0, stream>>>(agg, NH);
  scatter_kernel<<<cdiv((long)N_EDGES * 32, 256), B, 0, stream>>>(src, dst, dinv, hA, agg);
  post_kernel<<<cdiv(NH, 256), B, 0, stream>>>(agg, hA, dinv, b1, hB, 0xC0FFEE01u);

  // ---- layer 2 (K = 64)
  gemm_wmma<HID><<<N_NODES / 32, B, 0, stream>>>(hB, W2, hA);
  zero_f32<<<cdiv(NH, 256), B, 0, stream>>>(agg, NH);
  scatter_kernel<<<cdiv((long)N_EDGES * 32, 256), B, 0, stream>>>(src, dst, dinv, hA, agg);
  post_kernel<<<cdiv(NH, 256), B, 0, stream>>>(agg, hA, dinv, b2, hB, 0xC0FFEE02u);

  // ---- layer 3 (K = 64)
  gemm_wmma<HID><<<N_NODES / 32, B, 0, stream>>>(hB, W3, hA);
  zero_f32<<<cdiv(NH, 256), B, 0, stream>>>(agg, NH);
  scatter_kernel<<<cdiv((long)N_EDGES * 32, 256), B, 0, stream>>>(src, dst, dinv, hA, agg);
  post_kernel<<<cdiv(NH, 256), B, 0, stream>>>(agg, hA, dinv, b3, hB, 0xC0FFEE03u);

  // ---- pool + head
  zero_f32<<<cdiv(N_GRAPHS * HID + N_GRAPHS, 256), B, 0, stream>>>(pooled, N_GRAPHS * HID + N_GRAPHS);
  pool_kernel<<<cdiv(NH, 256), B, 0, stream>>>(hB, bat, pooled, cnt);
  head_kernel<<<1, B, 0, stream>>>(pooled, cnt, Wm1, bm1, Wm2, bm2, out, 0xC0FFEE04u);
}